// FullGNN_11192684773415
// MI455X (gfx1250) — compile-verified
//
#include <hip/hip_runtime.h>

#define NU    50000
#define NI    50000
#define NTOT  100000
#define DD    256
#define NNZ   400000
#define NLAYERS 2
#define ALPHA 0.2f

typedef float v2f __attribute__((ext_vector_type(2)));
typedef float v8f __attribute__((ext_vector_type(8)));

// ---------------------------------------------------------------------------
// SpMM: out[r, :] = sum_e in row r of (rows,cols,vals): vals[e] * ebs[cols[e], :]
// rows[] is sorted ascending -> binary-search the edge range per row.
// One 256-thread block per row; thread t owns column t (fully coalesced along D).
// Gathers of ebs rows hit the 192MB L2 (table is 102.4MB).
// ---------------------------------------------------------------------------
__global__ __launch_bounds__(256) void spmm_row_kernel(
    const int*   __restrict__ rows,
    const int*   __restrict__ cols,
    const float* __restrict__ vals,
    const float* __restrict__ ebs,   // [NTOT, DD]
    float*       __restrict__ out)   // [n_rows, DD]
{
    __shared__ int s_range[2];
    const int r = blockIdx.x;
    const int t = threadIdx.x;

    if (t == 0) {
        // lower_bound(r)
        int lo = 0, hi = NNZ;
        while (lo < hi) { int mid = (lo + hi) >> 1; if (rows[mid] < r) lo = mid + 1; else hi = mid; }
        s_range[0] = lo;
        // lower_bound(r+1), starting from previous result
        hi = NNZ;
        while (lo < hi) { int mid = (lo + hi) >> 1; if (rows[mid] < r + 1) lo = mid + 1; else hi = mid; }
        s_range[1] = lo;
    }
    __syncthreads();

    const int e0 = s_range[0];
    const int e1 = s_range[1];

    float acc = 0.0f;
    for (int e = e0; e < e1; ++e) {
        const float v = vals[e];          // uniform across wave -> broadcast
        const int   c = cols[e];
        acc = fmaf(v, ebs[(long)c * DD + t], acc);
    }
    out[(long)r * DD + t] = acc;
}

// ---------------------------------------------------------------------------
// Fused: out = leaky_relu( A_side @ W_side + (A_l * ebs_ent) @ W_dot, 0.2 )
// wave32 WMMA fp32: V_WMMA_F32_16X16X4_F32, K-loop over 256 in steps of 4.
// Block = 256 threads = 8 waves; block tile = 16 rows x 128 cols
// (each wave owns one 16x16 output tile). Grid = (n_rows/16, DD/128).
//
// ISA VGPR layouts used:
//   A (16x4 f32):  lane = m + 16*(k>>1), vgpr = k&1
//   B (4x16 f32):  lane = n + 16*(k>>1), vgpr = k&1
//   C/D (16x16):   vgpr j holds row (lane>>4)*8 + j, col = lane&15
// ---------------------------------------------------------------------------
__global__ __launch_bounds__(256) void fused_gemm_lrelu(
    const float* __restrict__ A_side,   // li  [n_rows, DD]
    const float* __restrict__ A_l,      // l   [n_rows, DD]
    const float* __restrict__ ebs_ent,  // ebs rows of this entity [n_rows, DD]
    const float* __restrict__ W_side,   // [DD, DD] (layer slice)
    const float* __restrict__ W_dot,    // [DD, DD] (layer slice)
    float*       __restrict__ out)      // [n_rows, DD]
{
    const int lane  = threadIdx.x & 31;
    const int wave  = threadIdx.x >> 5;           // 0..7
    const int m0    = blockIdx.x * 16;            // row tile origin
    const int n0    = blockIdx.y * 128 + wave * 16;

    const int sub   = lane & 15;                  // m for A loads, n for B loads
    const int khalf = lane >> 4;                  // selects K pair {0,1} vs {2,3}

    const float* aside_row = A_side  + (long)(m0 + sub) * DD;
    const float* al_row    = A_l     + (long)(m0 + sub) * DD;
    const float* ae_row    = ebs_ent + (long)(m0 + sub) * DD;

    v8f acc = {};

    #pragma unroll 4
    for (int k0 = 0; k0 < DD; k0 += 4) {
        const int ka = k0 + 2 * khalf;            // even -> 8B-aligned v2f loads

        v2f a1 = *(const v2f*)(aside_row + ka);
        v2f lv = *(const v2f*)(al_row + ka);
        v2f ev = *(const v2f*)(ae_row + ka);
        v2f a2 = lv * ev;                         // (l * ebs) elementwise

        const int nb = n0 + sub;
        v2f b1, b2;
        b1.x = W_side[(long)ka * DD + nb];
        b1.y = W_side[(long)(ka + 1) * DD + nb];
        b2.x = W_dot [(long)ka * DD + nb];
        b2.y = W_dot [(long)(ka + 1) * DD + nb];

        acc = __builtin_amdgcn_wmma_f32_16x16x4_f32(
                  false, a1, false, b1, (short)0, acc, false, false);
        acc = __builtin_amdgcn_wmma_f32_16x16x4_f32(
                  false, a2, false, b2, (short)0, acc, false, false);
    }

    // leaky_relu + store per C/D layout
    const int rbase = m0 + khalf * 8;
    const int cidx  = n0 + sub;
    #pragma unroll
    for (int j = 0; j < 8; ++j) {
        float x = acc[j];
        x = (x > 0.0f) ? x : ALPHA * x;
        out[(long)(rbase + j) * DD + cidx] = x;
    }
}

// ---------------------------------------------------------------------------
// Orchestration
// ---------------------------------------------------------------------------
extern "C" void kernel_launch(void* const* d_in, const int* in_sizes, int n_in,
                              void* d_out, int out_size, void* d_ws, size_t ws_size,
                              hipStream_t stream) {
    (void)in_sizes; (void)n_in; (void)out_size; (void)ws_size;

    const float* initial_ebs = (const float*)d_in[0];
    const float* W_side_u    = (const float*)d_in[1];
    const float* W_dot_u     = (const float*)d_in[2];
    const float* W_side_i    = (const float*)d_in[3];
    const float* W_dot_i     = (const float*)d_in[4];
    const float* LI_vals_u   = (const float*)d_in[5];
    const int*   LI_rows_u   = (const int*)  d_in[6];
    const int*   LI_cols_u   = (const int*)  d_in[7];
    const float* L_vals_u    = (const float*)d_in[8];
    const int*   L_rows_u    = (const int*)  d_in[9];
    const int*   L_cols_u    = (const int*)  d_in[10];
    const float* LI_vals_i   = (const float*)d_in[11];
    const int*   LI_rows_i   = (const int*)  d_in[12];
    const int*   LI_cols_i   = (const int*)  d_in[13];
    const float* L_vals_i    = (const float*)d_in[14];
    const int*   L_rows_i    = (const int*)  d_in[15];
    const int*   L_cols_i    = (const int*)  d_in[16];

    float* out = (float*)d_out;
    float* ws_li = (float*)d_ws;                    // [50000, 256]
    float* ws_l  = ws_li + (size_t)NU * DD;         // [50000, 256]

    const dim3 bs(256);
    const dim3 ggemm(NU / 16, DD / 128);            // 3125 x 2 (NU == NI)

    for (int l = 0; l < NLAYERS; ++l) {
        const float* ebs = (l == 0) ? initial_ebs
                                    : (out + (size_t)(l - 1) * NTOT * DD);
        float* fold_u = out + (size_t)l * NTOT * DD;
        float* fold_i = fold_u + (size_t)NU * DD;
        const size_t woff = (size_t)l * DD * DD;

        // entity 'u'
        spmm_row_kernel<<<NU, bs, 0, stream>>>(LI_rows_u, LI_cols_u, LI_vals_u, ebs, ws_li);
        spmm_row_kernel<<<NU, bs, 0, stream>>>(L_rows_u,  L_cols_u,  L_vals_u,  ebs, ws_l);
        fused_gemm_lrelu<<<ggemm, bs, 0, stream>>>(ws_li, ws_l, ebs,
                                                   W_side_u + woff, W_dot_u + woff, fold_u);
        // entity 'i'
        spmm_row_kernel<<<NI, bs, 0, stream>>>(LI_rows_i, LI_cols_i, LI_vals_i, ebs, ws_li);
        spmm_row_kernel<<<NI, bs, 0, stream>>>(L_rows_i,  L_cols_i,  L_vals_i,  ebs, ws_l);
        fused_gemm_lrelu<<<ggemm, bs, 0, stream>>>(ws_li, ws_l, ebs + (size_t)NU * DD,
                                                   W_side_i + woff, W_dot_i + woff, fold_i);
    }
}